// MCNet_37254546325976
// MI455X (gfx1250) — compile-verified
//
#include <hip/hip_runtime.h>

typedef __attribute__((ext_vector_type(2))) float v2f;
typedef __attribute__((ext_vector_type(8))) float v8f;

#define HT 128
#define WT 128
#define CCH 64

#define ACT_NONE 0
#define ACT_RELU 1
#define ACT_LRELU 2

// Per-channel LDS pitch: 3 rows * 66 cols = 198 used, padded to 208 so the
// +2*pitch offset between half-waves lands 32 banks away (conflict-free).
#define CPITCH 208

// ---------------------------------------------------------------------------
// Weight transpose: dst[o][t*Cin+ci] = src[o][ci*9+t]  (tap-major K order)
// ---------------------------------------------------------------------------
__global__ __launch_bounds__(256) void transpose_w(
    const float* __restrict__ src, float* __restrict__ dst, int O, int Cin)
{
    int K = Cin * 9;
    int i = blockIdx.x * 256 + threadIdx.x;
    if (i < O * K) {
        int o  = i / K;
        int r  = i - o * K;
        int t  = r / Cin;
        int ci = r - t * Cin;
        dst[i] = src[o * K + ci * 9 + t];
    }
}

// ---------------------------------------------------------------------------
// 3x3 conv ('SAME') as implicit GEMM, fp32 WMMA, tap-major K ordering.
// Block = 256 threads (8 wave32), 64-pixel strip of one row.
// Waves: mt = wave&3 (out-ch tile of 16), np = wave>>2 (pixel pair of 32).
// Each wave owns 2 N-tiles -> A fragment reused across 2 WMMAs.
// res / res2: optional tensors added after activation (residual, final sum).
// ---------------------------------------------------------------------------
__global__ __launch_bounds__(256) void conv3x3_wmma(
    const float* __restrict__ inA, const float* __restrict__ inB,
    int C0, int C1,
    const float* __restrict__ Wt,   // [64][9*Cin], tap-major (pre-transposed)
    const float* __restrict__ bias,
    const float* __restrict__ res,
    const float* __restrict__ res2,
    float* __restrict__ outp, int act)
{
    const int Cin = C0 + C1;
    const int K   = Cin * 9;

    __shared__ float s_tile[128 * CPITCH];   // [ci][dy*66 + rx], rx in [0,66)

    const int x0  = blockIdx.x * 64;
    const int y   = blockIdx.y;
    const int b   = blockIdx.z;
    const int tid = threadIdx.x;

    // -------- Stage zero-padded halo tile, division-free mapping. --------
    // lane l = tid&63 owns column rx = l (plus rx = l+64 for l < 2);
    // tid>>6 gives the starting channel, stepping by 4 (wave-uniform).
    {
        const int l = tid & 63;
        for (int ci = tid >> 6; ci < Cin; ci += 4) {
            const float* src = (ci < C0)
                ? inA + (size_t)(b * C0 + ci) * HT * WT
                : inB + (size_t)(b * C1 + (ci - C0)) * HT * WT;
            float* dstc = s_tile + ci * CPITCH;
            #pragma unroll
            for (int ry = 0; ry < 3; ++ry) {
                int gy = y + ry - 1;
                bool rowok = (gy >= 0) && (gy < HT);
                const float* srow = src + gy * WT;
                int gx = x0 + l - 1;
                float v = (rowok && gx >= 0 && gx < WT) ? srow[gx] : 0.f;
                dstc[ry * 66 + l] = v;
                if (l < 2) {                    // 2-column tail (66-wide row)
                    int gx2 = x0 + l + 63;      // always >= 63 >= 0
                    float v2 = (rowok && gx2 < WT) ? srow[gx2] : 0.f;
                    dstc[ry * 66 + l + 64] = v2;
                }
            }
        }
    }
    __syncthreads();

    const int wave = tid >> 5;
    const int lane = tid & 31;
    const int half = lane >> 4;
    const int lm   = lane & 15;
    const int mt   = wave & 3;
    const int np   = wave >> 2;
    const int colA = np * 32 + lm;       // pixel of N-tile 0
    const int colB = colA + 16;          // pixel of N-tile 1

    const float* wrow = Wt + (mt * 16 + lm) * K + 2 * half;
    const int cbase = 2 * half * CPITCH; // half-wave channel offset

    v8f acc0 = {};
    v8f acc1 = {};
    #pragma unroll
    for (int t = 0; t < 9; ++t) {
        const int tb = (t / 3) * 66 + (t % 3);
        const float* wt = wrow + t * Cin;
        const float* sb = s_tile + cbase + tb;
        #pragma unroll 4
        for (int ci = 0; ci < Cin; ci += 4) {
            v2f a;
            a.x = wt[ci];
            a.y = wt[ci + 1];
            const float* sc = sb + ci * CPITCH;
            v2f b0, b1;
            b0.x = sc[colA];
            b0.y = sc[CPITCH + colA];
            b1.x = sc[colB];
            b1.y = sc[CPITCH + colB];
            acc0 = __builtin_amdgcn_wmma_f32_16x16x4_f32(false, a, false, b0,
                                                         (short)0, acc0, false, false);
            acc1 = __builtin_amdgcn_wmma_f32_16x16x4_f32(false, a, false, b1,
                                                         (short)0, acc1, false, false);
        }
    }

    // Epilogue: C/D layout -> reg r, half h, lane lm = (M = r+8h, N = lm)
    #pragma unroll
    for (int r = 0; r < 8; ++r) {
        int o = mt * 16 + half * 8 + r;
        float bval = bias[o];
        float v0 = acc0[r] + bval;
        float v1 = acc1[r] + bval;
        if (act == ACT_RELU) {
            v0 = fmaxf(v0, 0.f);
            v1 = fmaxf(v1, 0.f);
        } else if (act == ACT_LRELU) {
            v0 = (v0 >= 0.f) ? v0 : 0.1f * v0;
            v1 = (v1 >= 0.f) ? v1 : 0.1f * v1;
        }
        int oiA = ((b * CCH + o) * HT + y) * WT + x0 + colA;
        int oiB = oiA + 16;
        if (res)  { v0 += res[oiA];  v1 += res[oiB]; }
        if (res2) { v0 += res2[oiA]; v1 += res2[oiB]; }
        outp[oiA] = v0;
        outp[oiB] = v1;
    }
}

// ---------------------------------------------------------------------------
// Deformable 3x3 conv (DG=8): bilinear gather -> LDS cols (tap-major rows),
// then fp32 WMMA GEMM with pre-transposed w_d. 32-pixel strip per block.
// ---------------------------------------------------------------------------
__global__ __launch_bounds__(256) void deform_conv_wmma(
    const float* __restrict__ refp, const float* __restrict__ offp,
    const float* __restrict__ Wt, const float* __restrict__ bias,
    float* __restrict__ outp)
{
    const int K = 576;
    __shared__ float s_cols[576 * 40];   // row stride 40

    const int x0  = blockIdx.x * 32;
    const int y   = blockIdx.y;
    const int b   = blockIdx.z;
    const int tid = threadIdx.x;
    const int HW  = HT * WT;

    // slots: n = tid&31 is the pixel; tid>>5 strides over (g,k) pairs.
    {
        const int n = tid & 31;
        const int x = x0 + n;
        for (int gk = tid >> 5; gk < 72; gk += 8) {
            int g = gk / 9;
            int k = gk - g * 9;

            const float* ob = offp +
                ((size_t)(b * 144 + gk * 2) * HT + y) * WT + x;
            float oy = ob[0];
            float ox = ob[HW];
            float py = (float)(y + (k / 3) - 1) + oy;
            float px = (float)(x + (k % 3) - 1) + ox;

            float y0f = floorf(py), x0f = floorf(px);
            float wy = py - y0f, wx = px - x0f;
            int yi0 = (int)y0f, xi0 = (int)x0f;
            int yi1 = yi0 + 1,  xi1 = xi0 + 1;

            float vy0 = (yi0 >= 0 && yi0 < HT) ? 1.f : 0.f;
            float vy1 = (yi1 >= 0 && yi1 < HT) ? 1.f : 0.f;
            float vx0 = (xi0 >= 0 && xi0 < WT) ? 1.f : 0.f;
            float vx1 = (xi1 >= 0 && xi1 < WT) ? 1.f : 0.f;
            int yc0 = min(max(yi0, 0), HT - 1);
            int yc1 = min(max(yi1, 0), HT - 1);
            int xc0 = min(max(xi0, 0), WT - 1);
            int xc1 = min(max(xi1, 0), WT - 1);

            float w00 = (1.f - wy) * (1.f - wx) * vy0 * vx0;
            float w01 = (1.f - wy) * wx         * vy0 * vx1;
            float w10 = wy         * (1.f - wx) * vy1 * vx0;
            float w11 = wy         * wx         * vy1 * vx1;
            int i00 = yc0 * WT + xc0;
            int i01 = yc0 * WT + xc1;
            int i10 = yc1 * WT + xc0;
            int i11 = yc1 * WT + xc1;

            const float* cb = refp + (size_t)(b * CCH + g * 8) * HW;
            #pragma unroll
            for (int c = 0; c < 8; ++c) {
                const float* p = cb + c * HW;
                float v = w00 * p[i00] + w01 * p[i01] +
                          w10 * p[i10] + w11 * p[i11];
                int kkrow = k * 64 + g * 8 + c;    // tap-major, matches Wt
                s_cols[kkrow * 40 + n] = v;
            }
        }
    }
    __syncthreads();

    const int wave = tid >> 5;
    const int lane = tid & 31;
    const int half = lane >> 4;
    const int lm   = lane & 15;
    const int mt   = wave & 3;
    const int nt   = wave >> 2;
    const int col  = nt * 16 + lm;
    const float* wrow = Wt + (mt * 16 + lm) * K + 2 * half;
    const float* sb = s_cols + 2 * half * 40 + col;

    v8f acc = {};
    #pragma unroll 4
    for (int kk = 0; kk < K; kk += 4) {
        v2f a;
        a.x = wrow[kk];
        a.y = wrow[kk + 1];
        const float* sc = sb + kk * 40;
        v2f bf;
        bf.x = sc[0];
        bf.y = sc[40];
        acc = __builtin_amdgcn_wmma_f32_16x16x4_f32(false, a, false, bf,
                                                    (short)0, acc, false, false);
    }

    #pragma unroll
    for (int r = 0; r < 8; ++r) {
        int o   = mt * 16 + half * 8 + r;
        float v = acc[r] + bias[o];
        v = (v >= 0.f) ? v : 0.1f * v;   // LReLU
        outp[((b * CCH + o) * HT + y) * WT + x0 + col] = v;
    }
}

// ---------------------------------------------------------------------------
extern "C" void kernel_launch(void* const* d_in, const int* in_sizes, int n_in,
                              void* d_out, int out_size, void* d_ws, size_t ws_size,
                              hipStream_t stream)
{
    const float* offset = (const float*)d_in[0];
    const float* ref    = (const float*)d_in[1];
    const float* w_d    = (const float*)d_in[2];
    const float* b_d    = (const float*)d_in[3];
    const float* w_c    = (const float*)d_in[4];
    const float* b_c    = (const float*)d_in[5];
    const float* rw1    = (const float*)d_in[6];
    const float* rb1    = (const float*)d_in[7];
    const float* rw2    = (const float*)d_in[8];
    const float* rb2    = (const float*)d_in[9];

    const int B = 4, C = 64, NB = 5;
    const size_t tn = (size_t)B * C * HT * WT;

    float* ws    = (float*)d_ws;
    float* t_out = ws;            // deform output ("out")
    float* t_h   = ws + tn;       // resblock hidden
    float* t_a   = ws + 2 * tn;   // out2 ping
    float* t_b   = ws + 3 * tn;   // out2 pong

    // Transposed (tap-major) weights live after the tensors.
    float* wt_d = ws + 4 * tn;                     // 64*576
    float* wt_c = wt_d + 64 * 576;                 // 64*1152
    float* wt_1 = wt_c + 64 * 1152;                // 5 * 64*576
    float* wt_2 = wt_1 + 5 * 64 * 576;             // 5 * 64*576

    dim3 blk(256);
    const int n576  = 64 * 576;
    const int n1152 = 64 * 1152;
    transpose_w<<<(n576 + 255) / 256, blk, 0, stream>>>(w_d, wt_d, 64, 64);
    transpose_w<<<(n1152 + 255) / 256, blk, 0, stream>>>(w_c, wt_c, 64, 128);
    for (int i = 0; i < NB; ++i) {
        transpose_w<<<(n576 + 255) / 256, blk, 0, stream>>>(
            rw1 + (size_t)i * n576, wt_1 + (size_t)i * n576, 64, 64);
        transpose_w<<<(n576 + 255) / 256, blk, 0, stream>>>(
            rw2 + (size_t)i * n576, wt_2 + (size_t)i * n576, 64, 64);
    }

    dim3 gridD(WT / 32, HT, B);   // deform: 32-px strips
    dim3 gridC(WT / 64, HT, B);   // conv:   64-px strips

    // out = lrelu(deform_conv(ref, offset, w_d, b_d))
    deform_conv_wmma<<<gridD, blk, 0, stream>>>(ref, offset, wt_d, b_d, t_out);

    // out2 = lrelu(conv3x3(concat(out, ref), w_c, b_c))
    conv3x3_wmma<<<gridC, blk, 0, stream>>>(t_out, ref, 64, 64, wt_c, b_c,
                                            nullptr, nullptr, t_a, ACT_LRELU);

    float* cur = t_a;
    float* nxt = t_b;
    for (int i = 0; i < NB; ++i) {
        conv3x3_wmma<<<gridC, blk, 0, stream>>>(
            cur, nullptr, 64, 0, wt_1 + (size_t)i * n576, rb1 + i * C,
            nullptr, nullptr, t_h, ACT_RELU);
        const bool last = (i == NB - 1);
        // Last conv2 also folds in the final "+ out" sum and writes d_out.
        conv3x3_wmma<<<gridC, blk, 0, stream>>>(
            t_h, nullptr, 64, 0, wt_2 + (size_t)i * n576, rb2 + i * C,
            cur, last ? t_out : nullptr,
            last ? (float*)d_out : nxt, ACT_NONE);
        float* tmp = cur; cur = nxt; nxt = tmp;
    }
}